// GLoRIALoss_19567871001344
// MI455X (gfx1250) — compile-verified
//
#include <hip/hip_runtime.h>
#include <math.h>

#define BB 48
#define CC 768
#define SS 361
#define SP 384      // padded S (24 row tiles of 16)
#define TT 64
#define KC 32       // K-chunk for GEMM1
#define CSTR 34     // LDS row stride (floats) for ctxT / wrdT, padded for banks
#define A2S 386     // LDS row stride (floats) for a2T
#define NEG_INF (-3.0e38f)

typedef float v2f __attribute__((ext_vector_type(2)));
typedef float v8f __attribute__((ext_vector_type(8)));

__device__ __forceinline__ v8f wmma_f32(v2f a, v2f b, v8f c) {
    // D = A(16x4 f32) * B(4x16 f32) + C(16x16 f32)
    return __builtin_amdgcn_wmma_f32_16x16x4_f32(false, a, false, b, (short)0, c,
                                                 false, false);
}

// ---------------------------------------------------------------------------
// Kernel 1: word norms  w1[c,t] = ||words[c,:,t]||   -> ws[0 .. 48*64)
// ---------------------------------------------------------------------------
__global__ void glo_w1_kernel(const float* __restrict__ wemb, float* __restrict__ ws) {
    int c = blockIdx.x, t = threadIdx.x;
    float s = 0.f;
    for (int d = 0; d < CC; d++) {
        float v = wemb[((size_t)c * CC + d) * TT + t];
        s += v * v;
    }
    ws[c * TT + t] = sqrtf(s);
}

// ---------------------------------------------------------------------------
// Kernel 2: fused per-(caption,image) pair pipeline.
// grid = 48*48 blocks, 256 threads (8 waves of 32).
// Writes agg[c,b] -> ws[3072 + c*48 + b], att_maps for diagonal pairs.
// ---------------------------------------------------------------------------
__global__ __launch_bounds__(256) void glo_pair_kernel(
    const float* __restrict__ lfeat,   // [48,768,361]
    const float* __restrict__ wemb,    // [48,768,64]
    const int*   __restrict__ caplens, // [48]
    float* __restrict__ ws,
    float* __restrict__ out)
{
    const int c    = blockIdx.x / BB;   // caption index
    const int b    = blockIdx.x % BB;   // image index
    const int tid  = threadIdx.x;
    const int lane = tid & 31;
    const int w    = tid >> 5;          // wave id 0..7
    const int h    = lane >> 4;         // lane half
    const int ln   = lane & 15;
    const int capc = caplens[c];

    const float* ctx_g = lfeat + (size_t)b * CC * SS;
    const float* wrd_g = wemb  + (size_t)c * CC * TT;

    extern __shared__ float sm[];
    float* ctxT = sm;                 // phase 1: [SP][CSTR]
    float* wrdT = sm + SP * CSTR;     // phase 1: [TT][CSTR]
    float* a2T  = sm;                 // phase 2: [TT][A2S] (aliases phase-1 space)
    __shared__ float redA[8][TT];
    __shared__ float redB[8][TT];
    __shared__ float fin[TT];

    // ================= GEMM1: scores[s,t] = sum_d ctx[d,s]*wrd[d,t] ==========
    // wave w owns row tiles rt = w, w+8, w+16 (24 tiles cover SP=384 rows).
    v8f acc[3][4] = {};

    for (int k0 = 0; k0 < CC; k0 += KC) {
        for (int idx = tid; idx < SP * KC; idx += 256) {
            int m = idx % SP, k = idx / SP;
            ctxT[m * CSTR + k] = (m < SS) ? ctx_g[(k0 + k) * SS + m] : 0.0f;
        }
        for (int idx = tid; idx < TT * KC; idx += 256) {
            int t = idx % TT, k = idx / TT;
            wrdT[t * CSTR + k] = wrd_g[(k0 + k) * TT + t];
        }
        __syncthreads();
        #pragma unroll
        for (int kk = 0; kk < KC; kk += 4) {
            v2f bf[4];
            #pragma unroll
            for (int ct = 0; ct < 4; ct++)
                bf[ct] = *(const v2f*)&wrdT[(ct * 16 + ln) * CSTR + kk + 2 * h];
            #pragma unroll
            for (int i = 0; i < 3; i++) {
                int rt = w + 8 * i;
                v2f af = *(const v2f*)&ctxT[(rt * 16 + ln) * CSTR + kk + 2 * h];
                #pragma unroll
                for (int ct = 0; ct < 4; ct++)
                    acc[i][ct] = wmma_f32(af, bf[ct], acc[i][ct]);
            }
        }
        __syncthreads();
    }

    // ============ softmax over words (T), masked; then z = a1*TEMP1 =========
    bool tval[4];
    #pragma unroll
    for (int ct = 0; ct < 4; ct++) tval[ct] = (ct * 16 + ln) < capc;

    #pragma unroll
    for (int i = 0; i < 3; i++) {
        #pragma unroll
        for (int v = 0; v < 8; v++) {
            float x[4];
            #pragma unroll
            for (int ct = 0; ct < 4; ct++) x[ct] = acc[i][ct][v];
            float mx = NEG_INF;
            #pragma unroll
            for (int ct = 0; ct < 4; ct++) mx = fmaxf(mx, tval[ct] ? x[ct] : NEG_INF);
            #pragma unroll
            for (int off = 8; off >= 1; off >>= 1) mx = fmaxf(mx, __shfl_xor(mx, off));
            float e[4], s = 0.f;
            #pragma unroll
            for (int ct = 0; ct < 4; ct++) {
                e[ct] = tval[ct] ? expf(x[ct] - mx) : 0.0f;
                s += e[ct];
            }
            #pragma unroll
            for (int off = 8; off >= 1; off >>= 1) s += __shfl_xor(s, off);
            float inv = 4.0f / s;   // TEMP1 folded in: z = a1 * 4
            #pragma unroll
            for (int ct = 0; ct < 4; ct++) acc[i][ct][v] = e[ct] * inv;
        }
    }

    // ========== softmax over pixels (S), rows >= 361 masked out ==============
    float cmp[4] = {NEG_INF, NEG_INF, NEG_INF, NEG_INF};
    #pragma unroll
    for (int i = 0; i < 3; i++) {
        int rt = w + 8 * i;
        #pragma unroll
        for (int v = 0; v < 8; v++) {
            int m = rt * 16 + v + 8 * h;
            bool ok = m < SS;
            #pragma unroll
            for (int ct = 0; ct < 4; ct++)
                cmp[ct] = fmaxf(cmp[ct], ok ? acc[i][ct][v] : NEG_INF);
        }
    }
    #pragma unroll
    for (int ct = 0; ct < 4; ct++) cmp[ct] = fmaxf(cmp[ct], __shfl_xor(cmp[ct], 16));
    if (lane < 16) {
        #pragma unroll
        for (int ct = 0; ct < 4; ct++) redA[w][ct * 16 + lane] = cmp[ct];
    }
    __syncthreads();
    float cmax[4];
    #pragma unroll
    for (int ct = 0; ct < 4; ct++) {
        float mx = redA[0][ct * 16 + ln];
        #pragma unroll
        for (int ww = 1; ww < 8; ww++) mx = fmaxf(mx, redA[ww][ct * 16 + ln]);
        cmax[ct] = mx;
    }
    float smp[4] = {0.f, 0.f, 0.f, 0.f};
    #pragma unroll
    for (int i = 0; i < 3; i++) {
        int rt = w + 8 * i;
        #pragma unroll
        for (int v = 0; v < 8; v++) {
            int m = rt * 16 + v + 8 * h;
            bool ok = m < SS;
            #pragma unroll
            for (int ct = 0; ct < 4; ct++) {
                float e = ok ? expf(acc[i][ct][v] - cmax[ct]) : 0.0f;
                acc[i][ct][v] = e;
                smp[ct] += e;
            }
        }
    }
    #pragma unroll
    for (int ct = 0; ct < 4; ct++) smp[ct] += __shfl_xor(smp[ct], 16);
    if (lane < 16) {
        #pragma unroll
        for (int ct = 0; ct < 4; ct++) redB[w][ct * 16 + lane] = smp[ct];
    }
    __syncthreads();
    float cinv[4];
    #pragma unroll
    for (int ct = 0; ct < 4; ct++) {
        float s = 0.f;
        #pragma unroll
        for (int ww = 0; ww < 8; ww++) s += redB[ww][ct * 16 + ln];
        cinv[ct] = 1.0f / s;
    }
    // store a2 transposed [t][s] into LDS (phase-1 buffers are dead)
    #pragma unroll
    for (int i = 0; i < 3; i++) {
        int rt = w + 8 * i;
        #pragma unroll
        for (int v = 0; v < 8; v++) {
            int m = rt * 16 + v + 8 * h;
            #pragma unroll
            for (int ct = 0; ct < 4; ct++)
                a2T[(ct * 16 + ln) * A2S + m] = acc[i][ct][v] * cinv[ct];
        }
    }
    __syncthreads();

    // diagonal pairs: dump attention maps  out[1 + b*T*S + t*S + s]
    if (c == b) {
        for (int idx = tid; idx < SS * TT; idx += 256) {
            int t = idx / SS, s = idx % SS;
            out[1 + (size_t)b * (TT * SS) + idx] = a2T[t * A2S + s];
        }
    }

    // ===== GEMM2: wei[d,t] = sum_s ctx[d,s]*a2[s,t]; fold into w12 / ||.||^2 =
    float w12p[4] = {0.f, 0.f, 0.f, 0.f};
    float sq2p[4] = {0.f, 0.f, 0.f, 0.f};
    for (int dti = 0; dti < 6; dti++) {
        int dt = w * 6 + dti;          // 48 d-tiles cover C=768
        v8f acc2[4] = {};
        for (int kk = 0; kk < 364; kk += 4) {
            int s0 = kk + 2 * h;
            int d  = dt * 16 + ln;     // A-matrix: lanes 0-15 and 16-31 both M=0..15
            v2f af;
            af.x = (s0     < SS) ? ctx_g[d * SS + s0]     : 0.0f;
            af.y = (s0 + 1 < SS) ? ctx_g[d * SS + s0 + 1] : 0.0f;
            #pragma unroll
            for (int ct = 0; ct < 4; ct++) {
                v2f bf = *(const v2f*)&a2T[(ct * 16 + ln) * A2S + kk + 2 * h];
                acc2[ct] = wmma_f32(af, bf, acc2[ct]);
            }
        }
        #pragma unroll
        for (int ct = 0; ct < 4; ct++) {
            int tg = ct * 16 + ln;
            #pragma unroll
            for (int v = 0; v < 8; v++) {
                int d = dt * 16 + v + 8 * h;
                float wv = wrd_g[d * TT + tg];
                float we = acc2[ct][v];
                w12p[ct] += wv * we;
                sq2p[ct] += we * we;
            }
        }
    }
    #pragma unroll
    for (int ct = 0; ct < 4; ct++) {
        w12p[ct] += __shfl_xor(w12p[ct], 16);
        sq2p[ct] += __shfl_xor(sq2p[ct], 16);
    }
    if (lane < 16) {
        #pragma unroll
        for (int ct = 0; ct < 4; ct++) {
            redA[w][ct * 16 + lane] = w12p[ct];
            redB[w][ct * 16 + lane] = sq2p[ct];
        }
    }
    __syncthreads();
    if (tid < TT) {
        float w12 = 0.f, sq2 = 0.f;
        #pragma unroll
        for (int ww = 0; ww < 8; ww++) { w12 += redA[ww][tid]; sq2 += redB[ww][tid]; }
        float w1v = ws[c * TT + tid];
        float sim = w12 / fmaxf(w1v * sqrtf(sq2), 1e-8f);
        fin[tid] = (tid < capc) ? expf(5.0f * sim) : 0.0f;   // TEMP2
    }
    __syncthreads();
    if (tid == 0) {
        float ssum = 0.f;
        for (int t = 0; t < TT; t++) ssum += fin[t];
        ws[BB * TT + c * BB + b] = logf(ssum);               // agg[c,b]
    }
}

// ---------------------------------------------------------------------------
// Kernel 3: global cosine loss + diagonal cross-entropies -> out[0]
// ---------------------------------------------------------------------------
__global__ __launch_bounds__(256) void glo_final_kernel(
    const float* __restrict__ gfeat, const float* __restrict__ semb,
    const float* __restrict__ ws, float* __restrict__ out)
{
    __shared__ float M1[BB * BB];   // local similarities [img][cap]
    __shared__ float M2[BB * BB];   // global scores      [img][cap]
    __shared__ float gn[BB], sn[BB], part[BB];
    int tid = threadIdx.x;

    if (tid < BB) {
        float s = 0.f;
        for (int k = 0; k < CC; k++) { float v = gfeat[tid * CC + k]; s += v * v; }
        gn[tid] = sqrtf(s);
    } else if (tid < 2 * BB) {
        int j = tid - BB; float s = 0.f;
        for (int k = 0; k < CC; k++) { float v = semb[j * CC + k]; s += v * v; }
        sn[j] = sqrtf(s);
    }
    const float* agg = ws + BB * TT;
    for (int p = tid; p < BB * BB; p += 256) {
        int i = p / BB, j = p % BB;
        float dot = 0.f;
        for (int k = 0; k < CC; k++) dot += gfeat[i * CC + k] * semb[j * CC + k];
        M2[p] = dot;
        M1[p] = agg[j * BB + i] * 10.0f;    // TEMP3, transposed
    }
    __syncthreads();
    for (int p = tid; p < BB * BB; p += 256) {
        int i = p / BB, j = p % BB;
        M2[p] = M2[p] / fmaxf(gn[i] * sn[j], 1e-8f) * 10.0f;
    }
    __syncthreads();
    if (tid < BB) {
        int i = tid;
        float tot = 0.f;
        const float* mats[2] = { M1, M2 };
        #pragma unroll
        for (int mi = 0; mi < 2; mi++) {
            const float* M = mats[mi];
            float mx = NEG_INF;
            for (int j = 0; j < BB; j++) mx = fmaxf(mx, M[i * BB + j]);
            float s = 0.f;
            for (int j = 0; j < BB; j++) s += expf(M[i * BB + j] - mx);
            tot += mx + logf(s) - M[i * BB + i];
            mx = NEG_INF;
            for (int j = 0; j < BB; j++) mx = fmaxf(mx, M[j * BB + i]);
            s = 0.f;
            for (int j = 0; j < BB; j++) s += expf(M[j * BB + i] - mx);
            tot += mx + logf(s) - M[i * BB + i];
        }
        part[i] = tot;
    }
    __syncthreads();
    if (tid == 0) {
        float L = 0.f;
        for (int i = 0; i < BB; i++) L += part[i];
        out[0] = L / (float)BB;
    }
}

extern "C" void kernel_launch(void* const* d_in, const int* in_sizes, int n_in,
                              void* d_out, int out_size, void* d_ws, size_t ws_size,
                              hipStream_t stream) {
    const float* gfeat = (const float*)d_in[0];   // [48,768]
    const float* lfeat = (const float*)d_in[1];   // [48,768,19,19]
    const float* wemb  = (const float*)d_in[2];   // [48,768,64]
    const float* semb  = (const float*)d_in[3];   // [48,768]
    const int*   caps  = (const int*)d_in[4];     // [48]
    float* out = (float*)d_out;
    float* ws  = (float*)d_ws;   // [0,3072): w1 norms, [3072,3072+2304): agg

    glo_w1_kernel<<<BB, TT, 0, stream>>>(wemb, ws);
    size_t smem = (size_t)TT * A2S * sizeof(float);   // 98816 B dynamic LDS
    glo_pair_kernel<<<BB * BB, 256, smem, stream>>>(lfeat, wemb, caps, ws, out);
    glo_final_kernel<<<1, 256, 0, stream>>>(gfeat, semb, ws, out);
}